// position_encoder_65335042507032
// MI455X (gfx1250) — compile-verified
//
#include <hip/hip_runtime.h>
#include <math.h>

// ---------------------------------------------------------------------------
// Positional-encoding add:  out[b,i,j] = x[b,i,j] + pos[i,j]
//   pos[i,j] = sin(i / 10000^((j+1)/D))  (j even)
//              cos(i / 10000^((j+1)/D))  (j odd)
// B=4, S=8192, D=1024, fp32.
//
// Roofline (MI455X): 128 MiB read + 128 MiB write = 256 MiB single pass
//   -> ~11.5 us @ 23.3 TB/s. Transcendental work (8.4M sin/cos, 2.1M powf
//   after 4x row amortization) fits under that, so the kernel is HBM-bound.
//
// CDNA5 specifics used:
//  * GLOBAL_LOAD_ASYNC_TO_LDS_B128 (ASYNCcnt) stages the x tile into LDS with
//    no destination VGPRs, so the wave computes sin/cos while the DMA flies;
//    a single s_wait_asynccnt 0 orders the LDS read-back (per-lane own data,
//    no barrier needed).
//  * Non-temporal B128 stores for the write-once output stream.
// ---------------------------------------------------------------------------

#define D_MODEL   1024
#define TPB       256          // 256 threads * 4 cols = 1024 = D_MODEL
#define ROWS      4            // rows per block (amortizes powf 4x)
#define BATCH_MAX 4

typedef float v4f __attribute__((ext_vector_type(4)));

__global__ __launch_bounds__(TPB) void posenc_add_kernel(
    const float* __restrict__ x, float* __restrict__ out, int S, int B) {
  // 4 batches * 4 rows * 1024 floats = 64 KB LDS tile (<< 320 KB/WGP)
  __shared__ float tile[BATCH_MAX * ROWS * D_MODEL];

  const int t  = threadIdx.x;
  const int j0 = t * 4;                 // first of 4 contiguous columns
  const int i0 = blockIdx.x * ROWS;     // first row of this block's strip

  // ---- Issue CDNA5 async global->LDS staging for the whole x tile ---------
  for (int b = 0; b < B; ++b) {
#pragma unroll
    for (int r = 0; r < ROWS; ++r) {
      const float* gp = x + ((size_t)b * S + (size_t)(i0 + r)) * D_MODEL + j0;
      __attribute__((address_space(3))) float* lp =
          (__attribute__((address_space(3))) float*)
              &tile[(b * ROWS + r) * D_MODEL + j0];
      asm volatile("global_load_async_to_lds_b128 %0, %1, off"
                   :
                   : "v"(lp), "v"(gp)
                   : "memory");
    }
  }

  // ---- Compute pos values while the async loads are in flight -------------
  // dpow = 10000^((j+1)/D) depends only on j: compute once per thread (4 cols),
  // reuse across ROWS rows and all B batches.
  float dpow[4];
#pragma unroll
  for (int u = 0; u < 4; ++u) {
    const float e = (float)(j0 + u + 1) * (1.0f / (float)D_MODEL);
    dpow[u] = powf(10000.0f, e);
  }

  float pv[ROWS][4];
#pragma unroll
  for (int r = 0; r < ROWS; ++r) {
    const float fi = (float)(i0 + r);
#pragma unroll
    for (int u = 0; u < 4; ++u) {
      const float pf = fi / dpow[u];
      // j0 is a multiple of 4, so parity of (j0+u) == parity of u.
      pv[r][u] = ((u & 1) == 0) ? sinf(pf) : cosf(pf);
    }
  }

  // ---- Wait for the async staging to land in LDS --------------------------
  asm volatile("s_wait_asynccnt 0" ::: "memory");

  // ---- Add and stream out with non-temporal 128-bit stores ----------------
  for (int b = 0; b < B; ++b) {
#pragma unroll
    for (int r = 0; r < ROWS; ++r) {
      const v4f xv = *(const v4f*)&tile[(b * ROWS + r) * D_MODEL + j0];
      v4f o;
      o.x = xv.x + pv[r][0];
      o.y = xv.y + pv[r][1];
      o.z = xv.z + pv[r][2];
      o.w = xv.w + pv[r][3];
      v4f* op = (v4f*)(out + ((size_t)b * S + (size_t)(i0 + r)) * D_MODEL + j0);
      __builtin_nontemporal_store(o, op);
    }
  }
}

extern "C" void kernel_launch(void* const* d_in, const int* in_sizes, int n_in,
                              void* d_out, int out_size, void* d_ws, size_t ws_size,
                              hipStream_t stream) {
  (void)n_in; (void)out_size; (void)d_ws; (void)ws_size;

  const float* x  = (const float*)d_in[0];
  float*      out = (float*)d_out;

  const int S = 8192;
  const int D = D_MODEL;
  const long long total = (long long)in_sizes[0];
  int B = (int)(total / ((long long)S * D));
  if (B < 1) B = 1;
  if (B > BATCH_MAX) B = BATCH_MAX;

  dim3 grid(S / ROWS);   // 2048 blocks
  dim3 block(TPB);       // 256 threads = 8 waves (wave32)
  posenc_add_kernel<<<grid, block, 0, stream>>>(x, out, S, B);
}